// MeanAggregator_4114578670327
// MI455X (gfx1250) — compile-verified
//
#include <hip/hip_runtime.h>
#include <hip/hip_bf16.h>

// Problem constants (match the reference).
#define BB     2048
#define SS     32
#define HH     256
#define NSEG   (BB * SS)   // 65536
#define NNBR   524288      // multiple of 4 -> aligned-chunk loop never OOB
#define OUT3H  768         // 3*H concat width

typedef __attribute__((ext_vector_type(2))) float v2f;
typedef __attribute__((ext_vector_type(8))) float v8f;
typedef __attribute__((ext_vector_type(4))) int   v4i;
// Exact pointee type the async builtins expect: GCC-style int4 vector.
typedef int v4i_g __attribute__((vector_size(16)));
#define AS1_V4I(p) ((__attribute__((address_space(1))) v4i_g*)(p))
#define AS3_V4I(p) ((__attribute__((address_space(3))) v4i_g*)(p))

__device__ __forceinline__ int lower_bound_i32(const int* __restrict__ a, int n, int v) {
    int lo = 0, hi = n;
    while (lo < hi) {
        int mid = (lo + hi) >> 1;
        if (a[mid] < v) lo = mid + 1; else hi = mid;
    }
    return lo;
}

// ---------------------------------------------------------------------------
// Segment-mean via WMMA: one wave owns 16 consecutive segments and half of H
// (128 columns). Sorted segment_ids => the group's neighbors are a contiguous
// run [base, endr). The chunk loop is 4-aligned: elements outside [base,endr)
// belong to other groups, so seg[k] != target automatically zeroes their A
// entries -- no clamps, no divergence. Per K=4 chunk:
//   A (16x4 f32): A[m,k] = (segment_ids[chunk+k] == seg0+m)
//   B (4x16 f32): gathered embedding columns for the current 16-col tile
// accumulated with V_WMMA_F32_16X16X4_F32 (8 tiles -> 8 accs, 64 VGPRs).
// This is exactly the reference's sparse.mm(mask, embeds_stack); counts come
// free from the per-segment boundaries; mean = acc * (1/max(c,1)).
// ---------------------------------------------------------------------------
__global__ __launch_bounds__(32) void seg_mean_kernel(
    const int* __restrict__ nbr, const int* __restrict__ seg,
    const float* __restrict__ ent, float* __restrict__ out) {

    const int g    = (int)blockIdx.x >> 1;
    const int half = (int)blockIdx.x & 1;          // which 128 columns of H
    const int seg0 = g * 16;
    const int lane = (int)threadIdx.x;

    __shared__ int bnd[17];
    if (lane <= 16) bnd[lane] = lower_bound_i32(seg, NNBR, seg0 + lane);
    __syncthreads();
    const int base = bnd[0];
    const int endr = bnd[16];
    (void)base; (void)endr;

#if __has_builtin(__builtin_amdgcn_wmma_f32_16x16x4_f32)
    __shared__ float rinv[16];
    if (lane < 16) {
        float c = (float)(bnd[lane + 1] - bnd[lane]);
        rinv[lane] = 1.0f / fmaxf(c, 1.0f);
    }
    __syncthreads();

    const int l15    = lane & 15;
    const bool hiK   = (lane >= 16);               // lanes 16-31 own K=2,3
    const int target = seg0 + l15;                 // A-row this lane tests
    const float* entH = ent + half * 128 + l15;

    v8f acc[8] = {};                               // 8 N-tiles of 16 cols

    if (base < endr) {
        const int cBeg = base & ~3;                // aligned, in-bounds
        const int cEnd = (endr + 3) & ~3;          // NNBR % 4 == 0
        for (int c = cBeg; c < cEnd; c += 4) {
            // Wave-uniform 16B loads of the chunk's ids (scalarizable).
            v4i sv = *(const v4i*)(seg + c);
            v4i iv = *(const v4i*)(nbr + c);
            const int s0 = hiK ? sv.z : sv.x;      // branchless K-half select
            const int s1 = hiK ? sv.w : sv.y;
            const int i0 = hiK ? iv.z : iv.x;
            const int i1 = hiK ? iv.w : iv.y;
            v2f A;
            A.x = (s0 == target) ? 1.0f : 0.0f;    // auto-0 outside [base,endr)
            A.y = (s1 == target) ? 1.0f : 0.0f;
            const float* r0 = entH + (size_t)i0 * HH;
            const float* r1 = entH + (size_t)i1 * HH;
#pragma unroll
            for (int t = 0; t < 8; ++t) {
                v2f Bt;
                Bt.x = r0[t * 16];                 // lanes 0-15: 64B run (L2-hot)
                Bt.y = r1[t * 16];
                acc[t] = __builtin_amdgcn_wmma_f32_16x16x4_f32(
                    false, A, false, Bt, (short)0, acc[t], false, false);
            }
        }
    }

    // C/D layout: VGPR r, lanes 0-15 -> M=r ; lanes 16-31 -> M=8+r.
    const int mBase = hiK ? 8 : 0;
#pragma unroll
    for (int r = 0; r < 8; ++r) {
        const int   M   = mBase + r;
        const float inv = rinv[M];
        float* o = out + (size_t)(seg0 + M) * OUT3H + half * 128 + l15;
#pragma unroll
        for (int t = 0; t < 8; ++t) o[t * 16] = acc[t][r] * inv;
    }
#else
    // Host-pass / no-WMMA fallback (same launch shape): each lane owns 4 cols.
    const int c0 = half * 128 + lane * 4;
    for (int m = 0; m < 16; ++m) {
        int s = bnd[m], e = bnd[m + 1];
        float a0 = 0.f, a1 = 0.f, a2 = 0.f, a3 = 0.f;
        for (int i = s; i < e; ++i) {
            const float* r = ent + (size_t)nbr[i] * HH + c0;
            a0 += r[0]; a1 += r[1]; a2 += r[2]; a3 += r[3];
        }
        float inv = 1.0f / fmaxf((float)(e - s), 1.0f);
        float* o = out + (size_t)(seg0 + m) * OUT3H + c0;
        o[0] = a0 * inv; o[1] = a1 * inv; o[2] = a2 * inv; o[3] = a3 * inv;
    }
#endif
}

// ---------------------------------------------------------------------------
// Broadcast ent/rel rows across the 32 timesteps. Async path: stage the 2KB
// [ent_row | rel_row] pair in LDS once via global_load_async_to_lds_b128, then
// replicate it 32x straight from LDS with global_store_async_from_lds_b128
// (no VGPR round-trip). ASYNCcnt drains at s_endpgm's implicit wait-idle.
// ---------------------------------------------------------------------------
#define BCAST_ASYNC (__has_builtin(__builtin_amdgcn_global_load_async_to_lds_b128) && \
                     __has_builtin(__builtin_amdgcn_global_store_async_from_lds_b128))

__global__ __launch_bounds__(256) void bcast_kernel(
    const int* __restrict__ s_tem, const int* __restrict__ r_tem,
    const float* __restrict__ ent, const float* __restrict__ rel,
    float* __restrict__ out) {

    const int b   = (int)blockIdx.x;
    const int tid = (int)threadIdx.x;
    const float* erow = ent + (size_t)s_tem[b] * HH;
    const float* rrow = rel + (size_t)r_tem[b] * HH;

#if BCAST_ASYNC
    __shared__ float buf[512];                     // [ent_row | rel_row], 2KB
    if (tid < 128) {                               // 128 lanes x 16B = 2KB
        const float* src = (tid < 64) ? (erow + tid * 4) : (rrow + (tid - 64) * 4);
        __builtin_amdgcn_global_load_async_to_lds_b128(
            AS1_V4I(src), AS3_V4I(&buf[tid * 4]), 0, 0);
    }
#if __has_builtin(__builtin_amdgcn_s_wait_asynccnt)
    __builtin_amdgcn_s_wait_asynccnt(0);
#else
    asm volatile("s_wait_asynccnt 0" ::: "memory");
#endif
    __syncthreads();                               // loads visible to all waves

    // 32 timesteps x 2KB = 64KB = 4096 x 16B chunks; 256 threads x 16 iters.
#pragma unroll 4
    for (int it = 0; it < 16; ++it) {
        int linear = it * 256 + tid;
        int s   = linear >> 7;                     // timestep
        int off = linear & 127;                    // 16B chunk within 2KB
        float* dst = out + (size_t)(b * SS + s) * OUT3H + HH + off * 4;
        __builtin_amdgcn_global_store_async_from_lds_b128(
            AS1_V4I(dst), AS3_V4I(&buf[off * 4]), 0, 0);
    }
#else
    const float vE = erow[tid];
    const float vR = rrow[tid];
    float* o = out + (size_t)b * SS * OUT3H + HH + tid;
    for (int s = 0; s < SS; ++s) {
        o[0]  = vE;
        o[HH] = vR;
        o += OUT3H;
    }
#endif
}

extern "C" void kernel_launch(void* const* d_in, const int* in_sizes, int n_in,
                              void* d_out, int out_size, void* d_ws, size_t ws_size,
                              hipStream_t stream) {
    (void)in_sizes; (void)n_in; (void)out_size; (void)d_ws; (void)ws_size;
    const int*   nbr   = (const int*)d_in[0];
    const int*   seg   = (const int*)d_in[1];
    const int*   s_tem = (const int*)d_in[2];
    const int*   r_tem = (const int*)d_in[3];
    const float* dt    = (const float*)d_in[4];
    const float* ent   = (const float*)d_in[5];
    const float* rel   = (const float*)d_in[6];
    float* out = (float*)d_out;

    // 4096 groups of 16 segments x 2 column-halves -> 8192 single-wave blocks.
    seg_mean_kernel<<<dim3((NSEG / 16) * 2), dim3(32), 0, stream>>>(nbr, seg, ent, out);
    // One block per batch row for the ent/rel broadcast.
    bcast_kernel<<<dim3(BB), dim3(256), 0, stream>>>(s_tem, r_tem, ent, rel, out);
    // s_hist_dt_seq is a flat copy appended after the [B,S,3H] tensor.
    (void)hipMemcpyAsync(out + (size_t)BB * SS * OUT3H, dt,
                         (size_t)NSEG * sizeof(float),
                         hipMemcpyDeviceToDevice, stream);
}